// LQR_20109036880507
// MI455X (gfx1250) — compile-verified
//
#include <hip/hip_runtime.h>
#include <hip/hip_bf16.h>

// Problem constants (from reference)
#define NB  64
#define T_  128
#define NS  64
#define NC  32
#define NSC 96

typedef __attribute__((ext_vector_type(2))) float v2f;
typedef __attribute__((ext_vector_type(8))) float v8f;
typedef __attribute__((ext_vector_type(4))) unsigned int u32x4;
typedef __attribute__((ext_vector_type(8))) int i32x8;
typedef __attribute__((ext_vector_type(4))) int i32x4;

// ---------------------------------------------------------------------------
// WMMA helpers: V_WMMA_F32_16X16X4_F32, fp32 16x16 tile, K stepped by 4.
// Fragment layouts per CDNA5 ISA 7.12.2.
// ---------------------------------------------------------------------------
template <typename FA, typename FB>
__device__ __forceinline__ v8f gemm16x16(const FA& Ael, const FB& Bel,
                                         int m0, int n0, int K, v8f acc) {
  const int lane = threadIdx.x & 31;
  const int row  = lane & 15;
  const int koff = (lane >> 4) << 1;   // 0 for lanes 0-15, 2 for lanes 16-31
  for (int kk = 0; kk < K; kk += 4) {
    v2f a, b;
    a[0] = Ael(m0 + row, kk + koff);
    a[1] = Ael(m0 + row, kk + koff + 1);
    b[0] = Bel(kk + koff,     n0 + row);
    b[1] = Bel(kk + koff + 1, n0 + row);
    acc = __builtin_amdgcn_wmma_f32_16x16x4_f32(
        false, a, false, b, (short)0, acc, false, false);
  }
  return acc;
}

template <typename FC>
__device__ __forceinline__ v8f load_tile16(const FC& Cel, int m0, int n0) {
  const int lane = threadIdx.x & 31;
  const int col  = lane & 15;
  const int mh   = (lane >> 4) << 3;
  v8f acc;
#pragma unroll
  for (int r = 0; r < 8; ++r) acc[r] = Cel(m0 + r + mh, n0 + col);
  return acc;
}

template <typename FD>
__device__ __forceinline__ void store_tile16(const FD& Del, int m0, int n0, v8f acc) {
  const int lane = threadIdx.x & 31;
  const int col  = lane & 15;
  const int mh   = (lane >> 4) << 3;
#pragma unroll
  for (int r = 0; r < 8; ++r) Del(m0 + r + mh, n0 + col, acc[r]);
}

// LDS layout (floats), padded rows (+1) to avoid 64-bank conflicts.
#define OFF_F    0                         // 64 x 97  (F = [A | B])
#define OFF_V    (OFF_F   + 64 * 97)       // 64 x 65  (V, overwritten by Vn)
#define OFF_QT   (OFF_V   + 64 * 65)       // 96 x 97  (Qt)
#define OFF_FTV  (OFF_QT  + 96 * 97)       // 96 x 65  (FtV)
#define OFF_AUG  (OFF_FTV + 96 * 65)       // 32 x 65  ([Quu|I] Gauss-Jordan)
#define OFF_KT   (OFF_AUG + 32 * 65)       // 32 x 65  (Kt)
#define OFF_S    (OFF_KT  + 32 * 65)       // 32 x 65  (S = Qux + Quu*Kt)
#define OFF_VEC  (OFF_S   + 32 * 65)       // v(64) c1(64) qt(96) kt(32) tv(32)
#define OFF_QG   (OFF_VEC + 64 + 64 + 96 + 32 + 32)  // 2 x 96*96 TDM double buffer
#define SMEM_FLOATS (OFF_QG + 2 * NSC * NSC)

// ---------------------------------------------------------------------------
// Tensor Data Mover: DMA one 96x96 fp32 Q tile (36 KB, contiguous) into LDS.
// D# per CDNA5 ISA (group0/group1; groups 2-3 unused).
// This toolchain's builtin arity (probed): 6 args
//   (u32x4 g0, i32x8 g1, i32x4 g2, i32x4 g3, i32x8, i32 cpol).
// ---------------------------------------------------------------------------
__device__ __forceinline__ void tdm_load_q(const float* gsrc, unsigned lds_byte_off) {
  const unsigned long long ga = (unsigned long long)(uintptr_t)gsrc;
  const u32x4 g0 = {
    1u,                                            // count=1 (valid user descriptor)
    lds_byte_off,                                  // lds_addr (bytes)
    (unsigned)ga,                                  // global_addr[31:0]
    ((unsigned)(ga >> 32) & 0x1FFFFFFu) | 0x80000000u  // global_addr[56:32] | type=2
  };
  const i32x8 g1 = {
    (int)0x00020000,   // workgroup_mask=0, data_size=2 (4 bytes), no flags
    (int)0x24000000,   // bits 63:48 = tensor_dim0 low16 = 9216
    (int)0x00010000,   // bits 95:80 = tensor_dim1 low16 = 1
    (int)0x24000000,   // bits 127:112 = tile_dim0 = 9216
    0,                 // tile_dim1 = 0 (unused), tile_dim2 = 0
    9216,              // tensor_dim0_stride low 32
    0, 0
  };
  const i32x4 z4 = {0, 0, 0, 0};
  const i32x8 z8 = {0, 0, 0, 0, 0, 0, 0, 0};
  __builtin_amdgcn_tensor_load_to_lds(g0, g1, z4, z4, z8, 0);
}

// ---------------------------------------------------------------------------
// Backward Riccati sweep. One block (256 threads = 8 waves) per batch.
// ---------------------------------------------------------------------------
__global__ void lqr_backward(const float* __restrict__ Q, const float* __restrict__ p,
                             const float* __restrict__ A, const float* __restrict__ Bm,
                             const float* __restrict__ c1,
                             float* __restrict__ Kws, float* __restrict__ kws) {
  extern __shared__ float smem[];
  float* sF   = smem + OFF_F;
  float* sV   = smem + OFF_V;
  float* sQt  = smem + OFF_QT;
  float* sFtV = smem + OFF_FTV;
  float* sAug = smem + OFF_AUG;
  float* sKt  = smem + OFF_KT;
  float* sS   = smem + OFF_S;
  float* sv   = smem + OFF_VEC;
  float* sc1  = sv  + 64;
  float* sqt  = sc1 + 64;
  float* skt  = sqt + 96;
  float* stv  = skt + 32;

  const int b   = blockIdx.x;
  const int tid = threadIdx.x;
  // Wave id as an SGPR so tile loops compile as scalar control flow
  // (EXEC provably all-1s around every v_wmma, per ISA §7.12 requirement).
  const int wv  = __builtin_amdgcn_readfirstlane(tid >> 5);

  // Preload F = [A | B] (64 x 96) and c1; V = 0, v = 0.
  for (int idx = tid; idx < NS * NSC; idx += 256) {
    int j = idx / NSC, i = idx - j * NSC;
    sF[j * 97 + i] = (i < NS) ? A[((size_t)b * NS + j) * NS + i]
                              : Bm[((size_t)b * NS + j) * NC + (i - NS)];
  }
  for (int idx = tid; idx < NS * NS; idx += 256)
    sV[(idx >> 6) * 65 + (idx & 63)] = 0.0f;
  if (tid < NS) { sv[tid] = 0.0f; sc1[tid] = c1[(size_t)b * NS + tid]; }

  // Kick off the TDM for the first (t = T-1) Q tile.
  if (wv == 0)
    tdm_load_q(Q + ((size_t)(b * T_ + (T_ - 1))) * (NSC * NSC),
               (unsigned)((OFF_QG + ((T_ - 1) & 1) * NSC * NSC) * sizeof(float)));
  __syncthreads();

  for (int t = T_ - 1; t >= 0; --t) {
    const float* pg  = p + ((size_t)(b * T_ + t)) * NSC;
    float*       sQg = smem + OFF_QG + (t & 1) * NSC * NSC;

    // Wait for Q[t] in LDS; immediately queue the DMA for Q[t-1] so the
    // next transfer overlaps this entire step's compute.
    if (wv == 0) {
      __builtin_amdgcn_s_wait_tensorcnt(0);
      if (t > 0)
        tdm_load_q(Q + ((size_t)(b * T_ + (t - 1))) * (NSC * NSC),
                   (unsigned)((OFF_QG + ((t - 1) & 1) * NSC * NSC) * sizeof(float)));
    }
    __syncthreads();

    // (1) FtV = F^T * V   (96 x 64, K = 64): 24 tiles over 8 waves.
    for (int tile = wv; tile < 24; tile += 8) {
      const int tm = (tile >> 2) << 4, tn = (tile & 3) << 4;
      v8f acc = {0.f, 0.f, 0.f, 0.f, 0.f, 0.f, 0.f, 0.f};
      acc = gemm16x16([&](int m, int k) { return sF[k * 97 + m]; },
                      [&](int k, int n) { return sV[k * 65 + n]; }, tm, tn, 64, acc);
      store_tile16([&](int m, int n, float x) { sFtV[m * 65 + n] = x; }, tm, tn, acc);
    }
    __syncthreads();

    // (2) Qt = Q[t] + FtV * F   (96 x 96, K = 64): 36 tiles (C from LDS buffer).
    for (int tile = wv; tile < 36; tile += 8) {
      const int tm = (tile / 6) << 4, tn = (tile % 6) << 4;
      v8f acc = load_tile16([&](int m, int n) { return sQg[m * NSC + n]; }, tm, tn);
      acc = gemm16x16([&](int m, int k) { return sFtV[m * 65 + k]; },
                      [&](int k, int n) { return sF[k * 97 + n]; }, tm, tn, 64, acc);
      store_tile16([&](int m, int n, float x) { sQt[m * 97 + n] = x; }, tm, tn, acc);
    }
    __syncthreads();

    // (3) qt = p[t] + F^T v + FtV c1.
    if (tid < NSC) {
      float acc = pg[tid];
      for (int j = 0; j < NS; ++j) acc += sF[j * 97 + tid] * sv[j];
      for (int j = 0; j < NS; ++j) acc += sFtV[tid * 65 + j] * sc1[j];
      sqt[tid] = acc;
    }
    __syncthreads();

    // (4) Gauss-Jordan inverse of Quu (32x32 SPD): [Quu | I] -> [I | Quu^-1].
    {
      const int r = tid >> 3;      // row 0..31
      const int g = (tid & 7) * 8; // 8-col group
      for (int cc = g; cc < g + 8; ++cc)
        sAug[r * 65 + cc] = (cc < 32) ? sQt[(64 + r) * 97 + 64 + cc]
                                      : ((cc - 32) == r ? 1.0f : 0.0f);
      __syncthreads();
      for (int k = 0; k < 32; ++k) {
        const float pv = sAug[k * 65 + k];
        const float f  = sAug[r * 65 + k];
        const float pi = 1.0f / pv;
        __syncthreads();
        if (r == k)
          for (int cc = g; cc < g + 8; ++cc) sAug[k * 65 + cc] *= pi;
        __syncthreads();
        if (r != k)
          for (int cc = g; cc < g + 8; ++cc) sAug[r * 65 + cc] -= f * sAug[k * 65 + cc];
        __syncthreads();
      }
    }

    // (5) Kt = -(Quu^-1 * Qux)  (32 x 64, K = 32); kt = -(Quu^-1 * qu).
    for (int tile = wv; tile < 8; tile += 8) {
      const int tm = (tile >> 2) << 4, tn = (tile & 3) << 4;
      v8f acc = {0.f, 0.f, 0.f, 0.f, 0.f, 0.f, 0.f, 0.f};
      acc = gemm16x16([&](int m, int k) { return -sAug[m * 65 + 32 + k]; },
                      [&](int k, int n) { return sQt[(64 + k) * 97 + n]; }, tm, tn, 32, acc);
      store_tile16([&](int m, int n, float x) { sKt[m * 65 + n] = x; }, tm, tn, acc);
    }
    if (tid < NC) {
      float acc = 0.0f;
      for (int j = 0; j < NC; ++j) acc -= sAug[tid * 65 + 32 + j] * sqt[64 + j];
      skt[tid] = acc;
    }
    __syncthreads();

    // (6) S = Qux + Quu * Kt  (32 x 64, K = 32);  tv = qu + Quu * kt.
    for (int tile = wv; tile < 8; tile += 8) {
      const int tm = (tile >> 2) << 4, tn = (tile & 3) << 4;
      v8f acc = load_tile16([&](int m, int n) { return sQt[(64 + m) * 97 + n]; }, tm, tn);
      acc = gemm16x16([&](int m, int k) { return sQt[(64 + m) * 97 + 64 + k]; },
                      [&](int k, int n) { return sKt[k * 65 + n]; }, tm, tn, 32, acc);
      store_tile16([&](int m, int n, float x) { sS[m * 65 + n] = x; }, tm, tn, acc);
    }
    if (tid < NC) {
      float acc = sqt[64 + tid];
      for (int l = 0; l < NC; ++l) acc += sQt[(64 + tid) * 97 + 64 + l] * skt[l];
      stv[tid] = acc;
    }
    __syncthreads();

    // (7) Vn = Qxx + Qxu*Kt + Kt^T*S -> sV;  vn = qx + Qxu*kt + Kt^T*tv -> sv.
    for (int tile = wv; tile < 16; tile += 8) {
      const int tm = (tile >> 2) << 4, tn = (tile & 3) << 4;
      v8f acc = load_tile16([&](int m, int n) { return sQt[m * 97 + n]; }, tm, tn);
      acc = gemm16x16([&](int m, int k) { return sQt[m * 97 + 64 + k]; },
                      [&](int k, int n) { return sKt[k * 65 + n]; }, tm, tn, 32, acc);
      acc = gemm16x16([&](int m, int k) { return sKt[k * 65 + m]; },
                      [&](int k, int n) { return sS[k * 65 + n]; }, tm, tn, 32, acc);
      store_tile16([&](int m, int n, float x) { sV[m * 65 + n] = x; }, tm, tn, acc);
    }
    if (tid < NS) {
      float acc = sqt[tid];
      for (int j = 0; j < NC; ++j) acc += sQt[tid * 97 + 64 + j] * skt[j];
      for (int j = 0; j < NC; ++j) acc += sKt[j * 65 + tid] * stv[j];
      sv[tid] = acc;
    }

    // (8) Spill Kt, kt to workspace for the forward pass.
    const size_t kbase = ((size_t)(b * T_ + t)) * NC * NS;
    for (int idx = tid; idx < NC * NS; idx += 256)
      Kws[kbase + idx] = sKt[(idx >> 6) * 65 + (idx & 63)];
    if (tid < NC) kws[((size_t)(b * T_ + t)) * NC + tid] = skt[tid];
    __syncthreads();
  }
}

// ---------------------------------------------------------------------------
// Forward rollout + cost. One block (128 threads) per batch; pure VALU mat-vec.
// ---------------------------------------------------------------------------
__global__ void lqr_forward(const float* __restrict__ Q, const float* __restrict__ p,
                            const float* __restrict__ A, const float* __restrict__ Bm,
                            const float* __restrict__ c1, const float* __restrict__ x0,
                            const float* __restrict__ Kws, const float* __restrict__ kws,
                            float* __restrict__ outx, float* __restrict__ outu,
                            float* __restrict__ outc) {
  __shared__ float sAB[NS * 97];
  __shared__ float sKt[NC * 65];
  __shared__ float sx[NS], su[NC], sxu[NSC], sred[NSC];

  const int b = blockIdx.x, tid = threadIdx.x;
  for (int idx = tid; idx < NS * NSC; idx += 128) {
    int j = idx / NSC, i = idx - j * NSC;
    sAB[j * 97 + i] = (i < NS) ? A[((size_t)b * NS + j) * NS + i]
                               : Bm[((size_t)b * NS + j) * NC + (i - NS)];
  }
  if (tid < NS) sx[tid] = x0[(size_t)b * NS + tid];
  float cost = 0.0f;
  __syncthreads();

  for (int t = 0; t < T_; ++t) {
    const float* Qg = Q + ((size_t)(b * T_ + t)) * (NSC * NSC);
    const float* pg = p + ((size_t)(b * T_ + t)) * NSC;
    const size_t kb = ((size_t)(b * T_ + t)) * NC * NS;

    for (int idx = tid; idx < NC * NS; idx += 128)
      sKt[(idx >> 6) * 65 + (idx & 63)] = Kws[kb + idx];
    __syncthreads();

    if (tid < NC) {               // u = Kt x + kt
      float acc = kws[((size_t)(b * T_ + t)) * NC + tid];
      for (int j = 0; j < NS; ++j) acc += sKt[tid * 65 + j] * sx[j];
      su[tid] = acc;
    }
    __syncthreads();

    if (tid < NS) { outx[((size_t)(b * T_ + t)) * NS + tid] = sx[tid]; sxu[tid] = sx[tid]; }
    if (tid < NC) { outu[((size_t)(b * T_ + t)) * NC + tid] = su[tid]; sxu[NS + tid] = su[tid]; }
    __syncthreads();

    if (tid < NSC) {              // cost: 0.5 xu^T Q xu + xu.p
      float r = 0.0f;
      for (int j = 0; j < NSC; ++j) r += Qg[tid * NSC + j] * sxu[j];
      sred[tid] = 0.5f * sxu[tid] * r + sxu[tid] * pg[tid];
    }
    float xn = 0.0f;
    if (tid < NS) {               // x' = [A|B] xu + c1
      xn = c1[(size_t)b * NS + tid];
      for (int j = 0; j < NSC; ++j) xn += sAB[tid * 97 + j] * sxu[j];
    }
    __syncthreads();
    if (tid == 0) { float s = 0.0f; for (int i = 0; i < NSC; ++i) s += sred[i]; cost += s; }
    if (tid < NS) sx[tid] = xn;
    __syncthreads();
  }
  if (tid == 0) outc[b] = cost;
}

extern "C" void kernel_launch(void* const* d_in, const int* in_sizes, int n_in,
                              void* d_out, int out_size, void* d_ws, size_t ws_size,
                              hipStream_t stream) {
  const float* Q  = (const float*)d_in[0];
  const float* p  = (const float*)d_in[1];
  const float* A  = (const float*)d_in[2];
  const float* Bm = (const float*)d_in[3];
  const float* c1 = (const float*)d_in[4];
  const float* x0 = (const float*)d_in[5];

  // Workspace: K (NB*T*NC*NS floats = 64 MB) then k (NB*T*NC floats = 1 MB).
  float* Kws = (float*)d_ws;
  float* kws = Kws + (size_t)NB * T_ * NC * NS;

  float* outx = (float*)d_out;                      // (NB, T, NS)
  float* outu = outx + (size_t)NB * T_ * NS;        // (NB, T, NC)
  float* outc = outu + (size_t)NB * T_ * NC;        // (NB,)

  const size_t smem = (size_t)SMEM_FLOATS * sizeof(float);  // ~203 KB of the WGP's 320 KB LDS
  (void)hipFuncSetAttribute((const void*)lqr_backward,
                            hipFuncAttributeMaxDynamicSharedMemorySize, (int)smem);

  lqr_backward<<<NB, 256, smem, stream>>>(Q, p, A, Bm, c1, Kws, kws);
  lqr_forward<<<NB, 128, 0, stream>>>(Q, p, A, Bm, c1, x0, Kws, kws, outx, outu, outc);
}